// SplineNetwork_82334523064366
// MI455X (gfx1250) — compile-verified
//
#include <hip/hip_runtime.h>
#include <math.h>

// SplineNetwork on MI455X (gfx1250, wave32).
//
// Exact algebraic reduction of the reference: cubic_conv has support
// dist < 2h = 1/64, but the grid spacing is 2/127 > 1/64, so only the 4
// corners of the query's containing cell can contribute, and those corners
// are always within the 9-NN set. One lane per query, 4 gathers into the
// L2-resident tables, ~60 VALU ops. This is launch-latency bound; WMMA
// (GEMM distance matrix + top-k) would be ~1e5x more work.

#define NGRID 128

__device__ __forceinline__ float cubic_conv1(float a) {
    // strict inequalities match the reference masks (a==0,1,2 -> 0)
    float a2 = a * a;
    float a3 = a2 * a;
    float r1 =  1.5f * a3 - 2.5f * a2 + 1.0f;
    float r2 = -0.5f * a3 + 2.5f * a2 - 4.0f * a + 2.0f;
    float r = 0.0f;
    if (a > 0.0f && a < 1.0f)      r = r1;
    else if (a > 1.0f && a < 2.0f) r = r2;
    return r;
}

__global__ void __launch_bounds__(256)
spline_cell_kernel(const float2* __restrict__ x,
                   const float*  __restrict__ w,
                   const float2* __restrict__ cp,
                   float*        __restrict__ out,
                   int batch) {
    int tid = blockIdx.x * blockDim.x + threadIdx.x;

    // Warm GL2 for the gather tables at 128B stride while the address math
    // below is in flight. Lowers to global_prefetch_b8 (speculative, no
    // LOADcnt cost) on gfx1250.
    if (tid < NGRID * NGRID) {
        if ((tid & 31) == 0) __builtin_prefetch(&w[tid],  0, 1);  // 64 KB table
        if ((tid & 15) == 0) __builtin_prefetch(&cp[tid], 0, 1);  // 128 KB table
    }

    if (tid >= batch) return;

    float2 q = x[tid];

    // Containing cell: t[i] = -1 + i*(2/127)  =>  i = floor((x+1)*63.5).
    // Off-by-one at a grid line is harmless: the ambiguous column/row is the
    // only one with nonzero support and it belongs to both candidate cells.
    int ci = (int)floorf((q.x + 1.0f) * 63.5f);
    int cj = (int)floorf((q.y + 1.0f) * 63.5f);
    ci = min(max(ci, 0), NGRID - 2);
    cj = min(max(cj, 0), NGRID - 2);
    int base = ci * NGRID + cj;

    float acc = 0.0f;
#pragma unroll
    for (int di = 0; di < 2; ++di) {
#pragma unroll
        for (int dj = 0; dj < 2; ++dj) {
            int idx = base + di * NGRID + dj;
            float2 c = cp[idx];                 // gather (L2-resident)
            float wt = w[idx];                  // gather (L2-resident)
            float dx = q.x - c.x;
            float dy = q.y - c.y;
            float d  = sqrtf(dx * dx + dy * dy);
            // s = norm / h with h = 1/128 exact => multiply by 128 is exact
            acc = fmaf(wt, cubic_conv1(d * 128.0f), acc);
        }
    }
    out[tid] = acc;
}

extern "C" void kernel_launch(void* const* d_in, const int* in_sizes, int n_in,
                              void* d_out, int out_size, void* d_ws, size_t ws_size,
                              hipStream_t stream) {
    const float2* x  = (const float2*)d_in[0];   // (batch, 2) f32
    const float*  w  = (const float*) d_in[1];   // (N*N,)    f32
    const float2* cp = (const float2*)d_in[2];   // (N*N, 2)  f32
    float* out = (float*)d_out;                  // (batch,)  f32

    int batch = in_sizes[0] / 2;
    const int block = 256;                       // 8 wave32 waves per block
    int grid = (batch + block - 1) / block;

    hipLaunchKernelGGL(spline_cell_kernel, dim3(grid), dim3(block), 0, stream,
                       x, w, cp, out, batch);
}